// Decoder_36636071035490
// MI455X (gfx1250) — compile-verified
//
#include <hip/hip_runtime.h>
#include <math.h>

typedef __attribute__((ext_vector_type(2))) float v2f;
typedef __attribute__((ext_vector_type(4))) float v4f;
typedef __attribute__((ext_vector_type(8))) float v8f;

#define WORKER_NUM 2000
#define TASK_NUM   5000
#define ABILITY_NUM 64
#define EDGE_TYPE  10
#define INPUT_DIM  64
#define ROW        (TASK_NUM * EDGE_TYPE)   /* 50000 floats per worker row */
#define WPB        8                        /* workers per block in stream kernel */

// ---------------------------------------------------------------------------
// Kernel 1: gather tau = inputs[2000:, :10] into a contiguous 50000-f32 buffer
// (strided source: 10 useful columns out of 64). 200 KB -> lives in L2 after.
// ---------------------------------------------------------------------------
__global__ __launch_bounds__(256) void tau_gather(const float* __restrict__ in,
                                                  float* __restrict__ tau) {
    int i = blockIdx.x * 256 + threadIdx.x;
    if (i < ROW) {
        int t = i / EDGE_TYPE;
        int e = i - t * EDGE_TYPE;
        tau[i] = in[(size_t)(WORKER_NUM + t) * INPUT_DIM + e];
    }
}

// ---------------------------------------------------------------------------
// Kernel 2: per-worker params via V_WMMA_F32_16X16X4_F32.
// One wave32 per 16 workers (2000 = 125 * 16).
// A tile = 16x4 f32 slice of worker_feature; B tile = 4x16 with W[k..k+3] in
// column 0, zeros elsewhere -> D column 0 accumulates worker_feature @ W.
// A layout (16x4 f32): lane L: row = L&15, VGPR0 = K=(L>>4)*2, VGPR1 = K+1.
// B layout (4x16 f32): lane L: col = L&15, VGPR0 = K=(L>>4)*2, VGPR1 = K+1.
// D column 0 ends up on lanes 0 (M=0..7 in c[0..7]) and 16 (M=8..15).
// Then p1 = sigmoid(d+b), store (log2 p2, log2 p1 - log2 p2) per worker.
// ---------------------------------------------------------------------------
__global__ __launch_bounds__(32) void worker_params(const float* __restrict__ in,
                                                    const float* __restrict__ W,
                                                    const float* __restrict__ b,
                                                    float2* __restrict__ prm) {
    const int lane = threadIdx.x;
    const int row  = lane & 15;
    const int koff = (lane >> 4) * 2;
    const bool col0 = (row == 0);
    const float* A = in + (size_t)(blockIdx.x * 16) * INPUT_DIM;

    v8f c = {0.f, 0.f, 0.f, 0.f, 0.f, 0.f, 0.f, 0.f};
    #pragma unroll
    for (int k = 0; k < ABILITY_NUM; k += 4) {
        v2f a, bm;
        a.x = A[row * INPUT_DIM + k + koff];
        a.y = A[row * INPUT_DIM + k + koff + 1];
        float w0 = W[k + koff];
        float w1 = W[k + koff + 1];
        bm.x = col0 ? w0 : 0.0f;   // v_cndmask: EXEC stays all-1s for WMMA
        bm.y = col0 ? w1 : 0.0f;
        c = __builtin_amdgcn_wmma_f32_16x16x4_f32(
                /*neg_a=*/false, a, /*neg_b=*/false, bm,
                /*c_mod=*/(short)0, c, /*reuse_a=*/false, /*reuse_b=*/false);
    }

    if (col0) {
        float bias = b[0];
        int base = blockIdx.x * 16 + (lane >> 4) * 8;
        #pragma unroll
        for (int r = 0; r < 8; ++r) {
            float x  = c[r] + bias;
            float p1 = 1.0f / (1.0f + expf(-x));
            float p2 = (1.0f - p1) * (1.0f / 9.0f);
            float l2 = log2f(p2);
            float ld = log2f(p1) - l2;
            prm[base + r] = make_float2(l2, ld);
        }
    }
}

// ---------------------------------------------------------------------------
// Kernel 3: stream the 400 MB output. Store-bandwidth bound (~17us floor at
// 23.3 TB/s). P[w, i] = exp2(fma(tau[i], ld_w, l2_w)). Each thread owns one
// tau float4 (regular load -> stays in L2, reused by all workers) and writes
// it for 8 workers with 128-bit NON-TEMPORAL stores so the 400 MB stream does
// not evict tau/params from L2.
// ---------------------------------------------------------------------------
__global__ __launch_bounds__(256) void pow_bcast(const float* __restrict__ tau,
                                                 const float2* __restrict__ prm,
                                                 float* __restrict__ out) {
    int i4 = blockIdx.x * 256 + threadIdx.x;     // float4 index within a row
    if (i4 >= ROW / 4) return;
    v4f t = *(const v4f*)(tau + (size_t)i4 * 4);
    int wbase = blockIdx.y * WPB;
    #pragma unroll
    for (int j = 0; j < WPB; ++j) {
        float2 p = prm[wbase + j];               // uniform -> scalarized loads
        v4f r;
        r.x = exp2f(fmaf(t.x, p.y, p.x));
        r.y = exp2f(fmaf(t.y, p.y, p.x));
        r.z = exp2f(fmaf(t.z, p.y, p.x));
        r.w = exp2f(fmaf(t.w, p.y, p.x));
        __builtin_nontemporal_store(
            r, (v4f*)(out + (size_t)(wbase + j) * ROW + (size_t)i4 * 4));
    }
}

// ---------------------------------------------------------------------------
extern "C" void kernel_launch(void* const* d_in, const int* in_sizes, int n_in,
                              void* d_out, int out_size, void* d_ws, size_t ws_size,
                              hipStream_t stream) {
    const float* in = (const float*)d_in[0];   // (7000, 64)
    const float* W  = (const float*)d_in[1];   // (64, 1)
    const float* b  = (const float*)d_in[2];   // (1,)
    float* out = (float*)d_out;                // (2000, 5000, 10)

    float*  tau = (float*)d_ws;                // 50000 f32 = 200 KB
    float2* prm = (float2*)(tau + ROW);        // 2000 float2 = 16 KB (8B aligned)

    tau_gather<<<(ROW + 255) / 256, 256, 0, stream>>>(in, tau);
    worker_params<<<WORKER_NUM / 16, 32, 0, stream>>>(in, W, b, prm);

    dim3 grid((ROW / 4 + 255) / 256, WORKER_NUM / WPB);  // 49 x 250
    pow_bcast<<<grid, 256, 0, stream>>>(tau, prm, out);
}